// RnnLayer_47682726920861
// MI455X (gfx1250) — compile-verified
//
#include <hip/hip_runtime.h>

// BiLSTM persistent kernel for MI455X (gfx1250, wave32, WMMA).
// z_t = [x_t ; h_{t-1}] @ [W ; U] + b  via V_WMMA_F32_16X16X4_F32 (fp32-exact).
// Grid = 128 blocks: [0,64) forward LSTM, [64,128) backward LSTM.
// 512 threads = 16 waves = 4 waves/SIMD for load-latency hiding; each wave
// owns one 16x16 (M-tile, gate) output tile.
// Each half synchronizes per timestep with an atomic-counter device barrier.
// The block's U-slice (64 cols x 1024 K) is staged transposed into LDS once
// via async global->LDS loads, so recurrent B-fragments come from ds_load_b64.

typedef __attribute__((ext_vector_type(2))) float v2f;
typedef __attribute__((ext_vector_type(8))) float v8f;

#define BATCH 64
#define TSEQ  256
#define DIN   1024
#define HDIM  1024
#define GDIM  4096      // 4*H
#define NBLK_HALF 64
#define THREADS 512
#define UTP   1026      // padded LDS pitch (floats) for Ut rows
#define ZLP   17        // padded LDS pitch (floats) for z tiles

// dynamic LDS: Ut[64][UTP] floats, then zl[4][BATCH][ZLP] floats
#define SMEM_FLOATS (64 * UTP + 4 * BATCH * ZLP)

extern __shared__ float smem[];

__global__ __launch_bounds__(THREADS, 1)
void bilstm_persistent(const float* __restrict__ x,
                       const float* __restrict__ Wf, const float* __restrict__ Uf,
                       const float* __restrict__ bf,
                       const float* __restrict__ Wb, const float* __restrict__ Ub,
                       const float* __restrict__ bb,
                       float* __restrict__ out,       // [B, T*2H]
                       float* __restrict__ hbuf,      // [2 dir][2 buf][B*H]
                       unsigned* __restrict__ cnt)    // [2]
{
    const int dir  = (blockIdx.x >= NBLK_HALF) ? 1 : 0;
    const int blk  = blockIdx.x - dir * NBLK_HALF;   // 0..63
    const int j0   = blk * 16;                       // h-column slice
    const int tid  = threadIdx.x;
    const int lane = tid & 31;
    const int wv   = tid >> 5;                       // 0..15
    const int m0   = (wv & 3) * 16;                  // batch-row tile base
    const int gt   = wv >> 2;                        // gate 0..3 (i,f,g,o)

    const float* W    = dir ? Wb : Wf;
    const float* U    = dir ? Ub : Uf;
    const float* bias = dir ? bb : bf;
    float*    hb0   = hbuf + (size_t)dir * (2 * BATCH * HDIM);
    unsigned* mycnt = cnt + dir;

    float* Ut = smem;                    // [64][UTP]  (transposed U slice)
    float* zl = smem + 64 * UTP;         // [4][BATCH][ZLP]

    // ---- one-time stage: Ut[c][k] = U[k][gate(c)*H + j0 + (c&15)] ----
    // Async global->LDS loads do the transpose in the addressing (ASYNCcnt).
    for (int idx = tid; idx < 64 * 1024; idx += THREADS) {
        const int c = idx & 63;              // LDS col 0..63 (gate*16 + j)
        const int k = idx >> 6;              // K row 0..1023
        const int gcol = (c >> 4) * HDIM + j0 + (c & 15);
        const float* gp = U + (size_t)k * GDIM + gcol;
        const unsigned ldsoff = (unsigned)(uintptr_t)(&Ut[c * UTP + k]);
        asm volatile("global_load_async_to_lds_b32 %0, %1, off"
                     :: "v"(ldsoff), "v"(gp) : "memory");
    }
    asm volatile("s_wait_asynccnt 0x0" ::: "memory");
    __syncthreads();

    // Per-lane WMMA fragment coordinates (ISA 7.12.2, f32 16x16x4):
    const int lrow = lane & 15;          // A: M within tile / B: N within tile
    const int lk2  = (lane >> 4) * 2;    // K sub-offset (0 or 2)

    const size_t xrowbase = (size_t)(m0 + lrow) * (TSEQ * DIN);
    const int colg = gt * HDIM + j0 + lrow;              // W column, this gate
    const float* L = Ut + (size_t)(gt * 16 + lrow) * UTP + lk2;  // LDS B row

    // Elementwise ownership: thread owns 2 fixed (b,j) cells for all steps.
    float creg[2];
    float bi_[2], bff_[2], bg_[2], bo_[2];
    int   eb_[2], ej_[2];
#pragma unroll
    for (int q = 0; q < 2; ++q) {
        int idx = tid * 2 + q;
        eb_[q] = idx >> 4;               // batch row 0..63
        ej_[q] = idx & 15;               // local h column 0..15
        creg[q] = 0.0f;
        bi_[q]  = bias[0 * HDIM + j0 + ej_[q]];
        bff_[q] = bias[1 * HDIM + j0 + ej_[q]];
        bg_[q]  = bias[2 * HDIM + j0 + ej_[q]];
        bo_[q]  = bias[3 * HDIM + j0 + ej_[q]];
    }

    for (int s = 0; s < TSEQ; ++s) {
        const int t = dir ? (TSEQ - 1 - s) : s;      // input & output timestep
        const float* hprev = hb0 + (size_t)(s & 1) * (BATCH * HDIM);
        float*       hnext = hb0 + (size_t)((s + 1) & 1) * (BATCH * HDIM);

        v8f acc = {0.f,0.f,0.f,0.f,0.f,0.f,0.f,0.f};

        // warm next timestep's x rows (global_prefetch_b8)
        if (s + 1 < TSEQ) {
            const int tn = dir ? (TSEQ - 2 - s) : (s + 1);
            const float* pn = x + (size_t)(m0 + lrow) * (TSEQ * DIN)
                                + (size_t)tn * DIN + (size_t)(lane >> 4) * 512;
            __builtin_prefetch(pn, 0, 1);
        }

        // ---- K part 1: input projection, k in [0,1024), weights W (L2) ----
        {
            const float* Ap = x + xrowbase + (size_t)t * DIN + lk2;
            const float* B0 = W + (size_t)lk2 * GDIM + colg;
#pragma unroll 4
            for (int kb = 0; kb < DIN; kb += 4) {
                v2f a = *(const v2f*)(Ap + kb);
                v2f b;
                b.x = B0[(size_t)kb * GDIM];
                b.y = B0[(size_t)(kb + 1) * GDIM];
                acc = __builtin_amdgcn_wmma_f32_16x16x4_f32(
                    false, a, false, b, (short)0, acc, false, false);
            }
        }

        // ---- K part 2: recurrence, weights U from LDS (ds_load_b64) ----
        {
            const float* Ap = hprev + (size_t)(m0 + lrow) * HDIM + lk2;
#pragma unroll 8
            for (int kb = 0; kb < HDIM; kb += 4) {
                v2f a = *(const v2f*)(Ap + kb);
                v2f b = *(const v2f*)(L + kb);
                acc = __builtin_amdgcn_wmma_f32_16x16x4_f32(
                    false, a, false, b, (short)0, acc, false, false);
            }
        }

        // ---- spill z tile to LDS (C/D layout: M = v + 8*(lane>=16), N = lane&15)
        {
            const int r = m0 + ((lane >> 4) << 3);
#pragma unroll
            for (int v = 0; v < 8; ++v) {
                zl[(gt * BATCH + r + v) * ZLP + lrow] = acc[v];
            }
        }
        __syncthreads();

        // ---- gates / state update / outputs ----
        const size_t outt = (size_t)t * (2 * HDIM) + (size_t)dir * HDIM + j0;
#pragma unroll
        for (int q = 0; q < 2; ++q) {
            const int b = eb_[q], j = ej_[q];
            float zi = zl[(0 * BATCH + b) * ZLP + j] + bi_[q];
            float zf = zl[(1 * BATCH + b) * ZLP + j] + bff_[q];
            float zg = zl[(2 * BATCH + b) * ZLP + j] + bg_[q];
            float zo = zl[(3 * BATCH + b) * ZLP + j] + bo_[q];
            float ig = 1.0f / (1.0f + __expf(-zi));
            float fg = 1.0f / (1.0f + __expf(-zf));
            float gg = tanhf(zg);
            float og = 1.0f / (1.0f + __expf(-zo));
            float c  = fg * creg[q] + ig * gg;
            creg[q]  = c;
            float h  = og * tanhf(c);
            hnext[(size_t)b * HDIM + j0 + j] = h;
            out[(size_t)b * (TSEQ * 2 * HDIM) + outt + j] = h;
        }

        // ---- device barrier across this direction's 64 blocks ----
        if (s != TSEQ - 1) {
            __threadfence();             // release this thread's h stores
            __syncthreads();
            if (tid == 0) {
                __hip_atomic_fetch_add(mycnt, 1u, __ATOMIC_RELEASE,
                                       __HIP_MEMORY_SCOPE_AGENT);
                const unsigned target = (unsigned)(NBLK_HALF * (s + 1));
                while (__hip_atomic_load(mycnt, __ATOMIC_ACQUIRE,
                                         __HIP_MEMORY_SCOPE_AGENT) < target) {
                    __builtin_amdgcn_s_sleep(1);
                }
            }
            __syncthreads();
            __threadfence();             // acquire: see other blocks' h stores
        } else {
            __syncthreads();
        }
    }
}

extern "C" void kernel_launch(void* const* d_in, const int* in_sizes, int n_in,
                              void* d_out, int out_size, void* d_ws, size_t ws_size,
                              hipStream_t stream) {
    const float* x  = (const float*)d_in[0];   // [64,256,1,1024]
    const float* Wf = (const float*)d_in[1];   // [1024,4096]
    const float* Uf = (const float*)d_in[2];   // [1024,4096]
    const float* bf = (const float*)d_in[3];   // [4096]
    const float* Wb = (const float*)d_in[4];
    const float* Ub = (const float*)d_in[5];
    const float* bb = (const float*)d_in[6];
    float* out = (float*)d_out;                // [64, 256*2048]

    // Workspace: [2 dir][2 buf][B*H] f32 h-broadcast buffers + 2 barrier counters.
    const size_t hbuf_bytes = (size_t)2 * 2 * BATCH * HDIM * sizeof(float);
    float*    hbuf = (float*)d_ws;
    unsigned* cnt  = (unsigned*)((char*)d_ws + hbuf_bytes);

    // Zero h0 state and barrier counters (graph-capture safe, replayed each call).
    hipMemsetAsync(d_ws, 0, hbuf_bytes + 2 * sizeof(unsigned), stream);

    const size_t smem_bytes = (size_t)SMEM_FLOATS * sizeof(float);
    bilstm_persistent<<<2 * NBLK_HALF, THREADS, smem_bytes, stream>>>(
        x, Wf, Uf, bf, Wb, Ub, bb, out, hbuf, cnt);
}